// Electron2MetaNodes_18820546691344
// MI455X (gfx1250) — compile-verified
//
#include <hip/hip_runtime.h>
#include <math.h>

typedef __attribute__((ext_vector_type(2))) float v2f;
typedef __attribute__((ext_vector_type(8))) float v8f;

#define N_ELEC   2048
#define N_ATOMS  512
#define FEAT     128
#define N_UP     1024
#define NUM_RBF  32
#define DIM      256
#define K2F      256              // 2*FEAT
#define TANH_GAIN (5.0f / 3.0f)

// Pass-1 tiling
#define TA        64              // atoms per block
#define GE        64              // electrons per block
#define A_TILES   (N_ATOMS / TA)  // 8
#define E_GROUPS  (N_ELEC / GE)   // 32
#define WAVES     8
#define E_PER_WV  (GE / WAVES)    // 8

// -------------------------------------------------------------------------
// Pass 1: single streaming pass over h_one_two (512 MB).
//   Produces: wsAtom [E_GROUPS][N_ATOMS][FEAT]  (partial atom sums per e-group)
//             wsElec [A_TILES ][N_ELEC ][FEAT]  (partial elec sums per a-tile)
// Deterministic: elec accumulation is wave-private registers; atom accumulation
// uses staggered exclusive LDS rows with a barrier every 8 steps.
// -------------------------------------------------------------------------
__global__ __launch_bounds__(256) void pass1_kernel(
    const float* __restrict__ h,        // [E][A][F]
    const float* __restrict__ r,        // [E][A]
    const float* __restrict__ mu,       // [32]
    const float* __restrict__ sigma,    // [32]
    float* __restrict__ wsAtom,
    float* __restrict__ wsElec)
{
    __shared__ float atomAcc[TA * FEAT];   // 32 KB

    const int tid  = threadIdx.x;
    const int w    = tid >> 5;
    const int lane = tid & 31;

    const int a_base = blockIdx.x * TA;
    const int e_base = blockIdx.y * GE;

    // Per-lane RBF constants (lane == rbf index k)
    const float mu_l = mu[lane];
    const float sg   = sigma[lane];
    const float sp   = (sg > 20.0f) ? sg : log1pf(__expf(sg));
    const float inv_sp = 1.0f / sp;

    // zero LDS accumulator
    for (int i = tid; i < TA * FEAT; i += 256) atomAcc[i] = 0.0f;

    float eacc[E_PER_WV][4];
#pragma unroll
    for (int i = 0; i < E_PER_WV; ++i)
#pragma unroll
        for (int j = 0; j < 4; ++j) eacc[i][j] = 0.0f;

    const int f0 = 4 * lane;

    for (int t = 0; t < TA; ++t) {
        if ((t & 7) == 0) __syncthreads();   // separates colliding LDS rows (dt==8)
        const int a_local = (8 * w + t) & (TA - 1);
        const int a = a_base + a_local;

        float atemp[4] = {0.0f, 0.0f, 0.0f, 0.0f};
#pragma unroll
        for (int i = 0; i < E_PER_WV; ++i) {
            const int e = e_base + 8 * w + i;
            // RBF mean: lane k computes its Gaussian; butterfly-sum across wave
            const float rv = r[(size_t)e * N_ATOMS + a];
            const float d  = rv - mu_l;
            float x = __expf(-(d * d) * inv_sp);
            x += __shfl_xor(x, 16, 32);
            x += __shfl_xor(x, 8, 32);
            x += __shfl_xor(x, 4, 32);
            x += __shfl_xor(x, 2, 32);
            x += __shfl_xor(x, 1, 32);
            const float rbf = x * (1.0f / 32.0f);

            const float4 hv = *(const float4*)(h + ((size_t)e * N_ATOMS + a) * FEAT + f0);
            const float v0 = hv.x * rbf, v1 = hv.y * rbf, v2 = hv.z * rbf, v3 = hv.w * rbf;
            atemp[0] += v0; atemp[1] += v1; atemp[2] += v2; atemp[3] += v3;
            eacc[i][0] += v0; eacc[i][1] += v1; eacc[i][2] += v2; eacc[i][3] += v3;
        }
        float* dst = &atomAcc[a_local * FEAT + f0];
        dst[0] += atemp[0]; dst[1] += atemp[1]; dst[2] += atemp[2]; dst[3] += atemp[3];
    }
    __syncthreads();

    // write atom partials: wsAtom[e_group][a][f]
    for (int i = tid; i < TA * FEAT; i += 256) {
        const int a_local = i >> 7;
        const int f = i & (FEAT - 1);
        wsAtom[((size_t)blockIdx.y * N_ATOMS + a_base + a_local) * FEAT + f] = atomAcc[i];
    }
    // write elec partials: wsElec[a_tile][e][f] (wave-exclusive rows)
#pragma unroll
    for (int i = 0; i < E_PER_WV; ++i) {
        const int e = e_base + 8 * w + i;
        float4 ev = make_float4(eacc[i][0], eacc[i][1], eacc[i][2], eacc[i][3]);
        *(float4*)(wsElec + ((size_t)blockIdx.x * N_ELEC + e) * FEAT + f0) = ev;
    }
}

// -------------------------------------------------------------------------
// Pass 2a: reduce atom partials over e-groups -> concat [A][2F] (up | dn), /1024
// -------------------------------------------------------------------------
__global__ __launch_bounds__(256) void atom_reduce_kernel(
    const float* __restrict__ wsAtom, float* __restrict__ concat)
{
    const int a    = blockIdx.x;
    const int half = threadIdx.x >> 7;       // 0 = up, 1 = dn
    const int f    = threadIdx.x & (FEAT - 1);
    float s = 0.0f;
#pragma unroll
    for (int g = 0; g < 16; ++g) {
        const int eg = half * 16 + g;
        s += wsAtom[((size_t)eg * N_ATOMS + a) * FEAT + f];
    }
    concat[(size_t)a * K2F + half * FEAT + f] = s * (1.0f / (float)N_UP);
}

// -------------------------------------------------------------------------
// Pass 2b: reduce elec partials over a-tiles -> d_out elec region, /512
// -------------------------------------------------------------------------
__global__ __launch_bounds__(256) void elec_reduce_kernel(
    const float* __restrict__ wsElec, float* __restrict__ out_elec)
{
    const int idx = blockIdx.x * 256 + threadIdx.x;   // e*F + f, 262144 total
    float s = 0.0f;
#pragma unroll
    for (int at = 0; at < A_TILES; ++at)
        s += wsElec[(size_t)at * N_ELEC * FEAT + idx];
    out_elec[idx] = s * (1.0f / (float)N_ATOMS);
}

// -------------------------------------------------------------------------
// Pass 3: atom_emb = tanh(concat @ W + b) * gain, via V_WMMA_F32_16X16X4_F32.
// One wave per 16x16 output tile; 64 k-steps of 4.
//   A 16x4 f32 layout: lanes 0-15 = M, {v0,v1} = K{0,1}; lanes 16-31 = K{2,3}
//   B 4x16  f32 layout: lanes = N, {v0,v1} = K{0,1} (lo half) / K{2,3} (hi half)
//   C/D v8f: vgpr j -> M = j (lanes 0-15) / M = j+8 (lanes 16-31), N = lane&15
// -------------------------------------------------------------------------
__global__ __launch_bounds__(128) void wmma_dense_kernel(
    const float* __restrict__ concat,   // [512][256]
    const float* __restrict__ W,        // [256][256]
    const float* __restrict__ bias,     // [256]
    float* __restrict__ out_atom)       // [512][256]
{
    const int wave = threadIdx.x >> 5;
    const int lane = threadIdx.x & 31;
    const int half = lane >> 4;
    const int lm   = lane & 15;

    const int tile  = blockIdx.x * 4 + wave;   // 512 tiles total
    const int mtile = tile >> 4;               // 0..31
    const int ntile = tile & 15;               // 0..15
    const int m0 = mtile * 16;
    const int n0 = ntile * 16;

    const float bcol = bias[n0 + lm];

    v8f acc = {};
    const float* arow = concat + (size_t)(m0 + lm) * K2F + half * 2;
#pragma unroll 4
    for (int k0 = 0; k0 < K2F; k0 += 4) {
        v2f av = *(const v2f*)(arow + k0);
        const int kb = k0 + half * 2;
        v2f bv;
        bv.x = W[(size_t)kb * DIM + n0 + lm];
        bv.y = W[(size_t)(kb + 1) * DIM + n0 + lm];
        acc = __builtin_amdgcn_wmma_f32_16x16x4_f32(
            /*neg_a=*/false, av, /*neg_b=*/false, bv,
            /*c_mod=*/(short)0, acc, /*reuse_a=*/false, /*reuse_b=*/false);
    }

#pragma unroll
    for (int j = 0; j < 8; ++j) {
        const int row = m0 + j + half * 8;
        const int col = n0 + lm;
        out_atom[(size_t)row * DIM + col] = tanhf(acc[j] + bcol) * TANH_GAIN;
    }
}

// -------------------------------------------------------------------------
extern "C" void kernel_launch(void* const* d_in, const int* in_sizes, int n_in,
                              void* d_out, int out_size, void* d_ws, size_t ws_size,
                              hipStream_t stream) {
    const float* h     = (const float*)d_in[0];
    const float* r     = (const float*)d_in[1];
    const float* mu    = (const float*)d_in[2];
    const float* sigma = (const float*)d_in[3];
    const float* W     = (const float*)d_in[4];
    const float* bias  = (const float*)d_in[5];

    float* out = (float*)d_out;
    float* out_atom = out;                       // 512*256 = 131072
    float* out_elec = out + N_ATOMS * DIM;       // 2048*128 = 262144

    float* ws      = (float*)d_ws;
    float* wsAtom  = ws;                                         // 32*512*128  = 2M floats
    float* wsElec  = wsAtom + (size_t)E_GROUPS * N_ATOMS * FEAT; // 8*2048*128  = 2M floats
    float* wsConc  = wsElec + (size_t)A_TILES * N_ELEC * FEAT;   // 512*256     = 128K floats

    dim3 g1(A_TILES, E_GROUPS);
    pass1_kernel<<<g1, 256, 0, stream>>>(h, r, mu, sigma, wsAtom, wsElec);
    atom_reduce_kernel<<<N_ATOMS, 256, 0, stream>>>(wsAtom, wsConc);
    elec_reduce_kernel<<<(N_ELEC * FEAT) / 256, 256, 0, stream>>>(wsElec, out_elec);
    wmma_dense_kernel<<<128, 128, 0, stream>>>(wsConc, W, bias, out_atom);
}